// DeepWave_38079180046691
// MI455X (gfx1250) — compile-verified
//
#include <hip/hip_runtime.h>
#include <hip/hip_bf16.h>
#include <math.h>

#define NPX 196608
#define NCH 64
#define BATCH 32
#define KDEG 5
#define DEG 10
#define NNZ (NPX * DEG)
#define N_UNROLL 5

typedef __attribute__((ext_vector_type(2))) float v2f;
typedef __attribute__((ext_vector_type(8))) float v8f;

// ---------------------------------------------------------------------------
// row_ptr[r] = lower_bound(rows, r)   (rows is sorted, length NNZ)
// ---------------------------------------------------------------------------
__global__ void rowptr_kernel(const int* __restrict__ rows, int* __restrict__ row_ptr) {
    int r = blockIdx.x * blockDim.x + threadIdx.x;
    if (r > NPX) return;
    int lo = 0, hi = NNZ;
    while (lo < hi) {
        int mid = (lo + hi) >> 1;
        if (rows[mid] < r) lo = mid + 1; else hi = mid;
    }
    row_ptr[r] = lo;
}

// ---------------------------------------------------------------------------
// (B,NPX) -> (NPX,B) pixel-major state
// ---------------------------------------------------------------------------
__global__ void transpose_in_kernel(const float* __restrict__ Iprev, float* __restrict__ X) {
    size_t i = (size_t)blockIdx.x * blockDim.x + threadIdx.x;
    if (i >= (size_t)NPX * BATCH) return;
    int b = (int)(i & (BATCH - 1));
    size_t p = i >> 5;
    X[i] = Iprev[(size_t)b * NPX + p];
}

// ---------------------------------------------------------------------------
// Back-projection: Y[p*32+b] = tau[p] * sum_{m,n} D[m,p] S[b,m,n] D[n,p]
// One block = 8 waves = one batch b, 8 consecutive 16-pixel tiles.
// S_b staged in LDS via GLOBAL_LOAD_ASYNC_TO_LDS_B128 (ASYNCcnt path);
// inner product via V_WMMA_F32_16X16X4_F32.
// ---------------------------------------------------------------------------
__global__ void __launch_bounds__(256)
backproj_kernel(const float* __restrict__ S, const float* __restrict__ D,
                const float* __restrict__ tau, float* __restrict__ Y) {
    __shared__ __align__(16) float sS[NCH * NCH];   // 16 KB
    const int b = blockIdx.y;
    const float* Sb = S + (size_t)b * NCH * NCH;

    // Async stage: 256 threads x 4 x B128 = 4096 floats, LDS written directly
    // (no VGPR bounce, tracked by ASYNCcnt instead of LOADcnt/DScnt).
#pragma unroll
    for (int it = 0; it < 4; ++it) {
        const int elt = threadIdx.x * 4 + it * 1024;          // float index, 16B chunks
        unsigned lds_off = (unsigned)(uintptr_t)(const void*)(sS + elt);
        unsigned long long ga = (unsigned long long)(uintptr_t)(Sb + elt);
        asm volatile("global_load_async_to_lds_b128 %0, %1, off"
                     :: "v"(lds_off), "v"(ga) : "memory");
    }
#if __has_builtin(__builtin_amdgcn_s_wait_asynccnt)
    __builtin_amdgcn_s_wait_asynccnt(0);
#else
    asm volatile("s_wait_asynccnt 0" ::: "memory");
#endif
    __syncthreads();

    const int wave = threadIdx.x >> 5;
    const int lane = threadIdx.x & 31;
    const int n    = lane & 15;       // column within tile / M row for A
    const int h    = lane >> 4;       // lane-half selector
    const int p0   = (blockIdx.x * 8 + wave) * 16;
    const int p    = p0 + n;

    // B operands: D[0:64, p0:p0+16]; lane holds rows {4kk+2h, 4kk+2h+1} at col n
    v2f Breg[16];
#pragma unroll
    for (int kk = 0; kk < 16; ++kk) {
        const int k0 = kk * 4 + 2 * h;
        Breg[kk].x = D[(size_t)k0 * NPX + p];
        Breg[kk].y = D[(size_t)(k0 + 1) * NPX + p];
    }

    float acc = 0.f;
#pragma unroll
    for (int mt = 0; mt < 4; ++mt) {
        v8f C = {};
        const int mrow = mt * 16 + n;     // A: M = lane&15
#pragma unroll
        for (int kk = 0; kk < 16; ++kk) {
            v2f A;
            A.x = sS[mrow * NCH + kk * 4 + 2 * h];
            A.y = sS[mrow * NCH + kk * 4 + 2 * h + 1];
            C = __builtin_amdgcn_wmma_f32_16x16x4_f32(
                    false, A, false, Breg[kk], (short)0, C, false, false);
        }
        // acc += sum_m D[m,p] * C[m, p-p0] for this M-tile
#pragma unroll
        for (int r = 0; r < 8; ++r) {
            const int mm = mt * 16 + r + 8 * h;   // C layout: M = r + 8*half
            acc += D[(size_t)mm * NPX + p] * C[r];
        }
    }
    // lanes l and l^16 hold complementary halves of the m-sum for the same p
    acc += __shfl_xor(acc, 16, 32);
    if (lane < 16) {
        Y[(size_t)p * BATCH + b] = tau[p] * acc;
    }
}

// ---------------------------------------------------------------------------
// Fused SpMM + Chebyshev term update. One wave per row, lane = batch index.
//   s       = (L @ Xg)[row, lane]
//   t       = Tprev2 ? 2*s - Tprev2[row] : s
//   Tnew    = t
//   Out     = init ? theta[0]*Xg[row] + theta[k]*t : Out + theta[k]*t
// ---------------------------------------------------------------------------
__global__ void __launch_bounds__(256)
spmm_cheb_kernel(const int* __restrict__ row_ptr, const int* __restrict__ cols,
                 const float* __restrict__ vals, const float* __restrict__ Xg,
                 const float* __restrict__ Tprev2, float* __restrict__ Tnew,
                 float* __restrict__ Out, const float* __restrict__ theta,
                 int k, int init) {
    const int wave = threadIdx.x >> 5;
    const int lane = threadIdx.x & 31;
    const int r = blockIdx.x * 8 + wave;
    if (r >= NPX) return;

    const int beg = row_ptr[r];
    const int end = row_ptr[r + 1];
    float s = 0.f;
    for (int i = beg; i < end; ++i) {
        const int   c = cols[i];
        const float v = vals[i];
        s = fmaf(v, Xg[(size_t)c * BATCH + lane], s);
    }
    const size_t idx = (size_t)r * BATCH + lane;
    const float t = Tprev2 ? (2.f * s - Tprev2[idx]) : s;
    Tnew[idx] = t;
    const float thk = theta[k];
    if (init) {
        Out[idx] = theta[0] * Xg[idx] + thk * t;
    } else {
        Out[idx] = Out[idx] + thk * t;
    }
}

// ---------------------------------------------------------------------------
// X = ReTanh(Out + Y); on final iteration also emit d_out in (B,NPX) layout
// ---------------------------------------------------------------------------
__global__ void retanh_kernel(const float* __restrict__ Out, const float* __restrict__ Y,
                              float* __restrict__ X, float* __restrict__ final_out,
                              int write_final) {
    size_t i = (size_t)blockIdx.x * blockDim.x + threadIdx.x;
    if (i >= (size_t)NPX * BATCH) return;
    float v = Out[i] + Y[i];
    v = v > 0.f ? v : 0.f;
    v = tanhf(v);          // ALPHA = 1.0
    X[i] = v;
    if (write_final) {
        int b = (int)(i & (BATCH - 1));
        size_t p = i >> 5;
        final_out[(size_t)b * NPX + p] = v;
    }
}

// ---------------------------------------------------------------------------
static inline size_t align_up(size_t x, size_t a) { return (x + a - 1) & ~(a - 1); }

extern "C" void kernel_launch(void* const* d_in, const int* in_sizes, int n_in,
                              void* d_out, int out_size, void* d_ws, size_t ws_size,
                              hipStream_t stream) {
    const float* S      = (const float*)d_in[0];
    const float* I_prev = (const float*)d_in[1];
    const int*   rows   = (const int*)  d_in[2];
    const int*   cols   = (const int*)  d_in[3];
    const float* vals   = (const float*)d_in[4];
    const float* Dm     = (const float*)d_in[5];
    const float* tau    = (const float*)d_in[6];
    const float* theta  = (const float*)d_in[7];
    float* out = (float*)d_out;

    const size_t state_bytes = (size_t)NPX * BATCH * sizeof(float);   // ~24 MB
    char* ws = (char*)d_ws;
    size_t off = 0;
    int*   row_ptr = (int*)(ws + off); off += align_up((size_t)(NPX + 1) * sizeof(int), 256);
    float* X   = (float*)(ws + off); off += align_up(state_bytes, 256);
    float* TA  = (float*)(ws + off); off += align_up(state_bytes, 256);
    float* TB  = (float*)(ws + off); off += align_up(state_bytes, 256);
    float* OUT = (float*)(ws + off); off += align_up(state_bytes, 256);
    float* Y   = (float*)(ws + off); off += align_up(state_bytes, 256);

    const size_t nelem = (size_t)NPX * BATCH;                 // 6 291 456
    const int eblocks  = (int)((nelem + 255) / 256);          // 24576
    const int rblocks  = NPX / 8;                             // one wave per row

    // 1) CSR row pointers from sorted rows[]
    rowptr_kernel<<<(NPX + 256) / 256, 256, 0, stream>>>(rows, row_ptr);

    // 2) transpose I_prev -> pixel-major state
    transpose_in_kernel<<<eblocks, 256, 0, stream>>>(I_prev, X);

    // 3) back-projection (WMMA fp32 + async LDS staging) -> Y (pixel-major)
    dim3 bg(NPX / 128, BATCH);
    backproj_kernel<<<bg, 256, 0, stream>>>(S, Dm, tau, Y);

    // 4) recurrent unroll: x = ReTanh(cheb(x) + y_proj)
    for (int it = 0; it < N_UNROLL; ++it) {
        // k=1: t1 = L x ; OUT = th0*x + th1*t1
        spmm_cheb_kernel<<<rblocks, 256, 0, stream>>>(row_ptr, cols, vals, X, nullptr,
                                                      TA, OUT, theta, 1, 1);
        // k=2: t2 = 2 L t1 - x ; OUT += th2*t2
        spmm_cheb_kernel<<<rblocks, 256, 0, stream>>>(row_ptr, cols, vals, TA, X,
                                                      TB, OUT, theta, 2, 0);
        // k=3: t3 = 2 L t2 - t1 ; OUT += th3*t3   (t3 overwrites TA; own-row RAW only)
        spmm_cheb_kernel<<<rblocks, 256, 0, stream>>>(row_ptr, cols, vals, TB, TA,
                                                      TA, OUT, theta, 3, 0);
        // k=4: t4 = 2 L t3 - t2 ; OUT += th4*t4
        spmm_cheb_kernel<<<rblocks, 256, 0, stream>>>(row_ptr, cols, vals, TA, TB,
                                                      TB, OUT, theta, 4, 0);
        // x = ReTanh(OUT + Y); last iter also writes d_out (B,NPX)
        retanh_kernel<<<eblocks, 256, 0, stream>>>(OUT, Y, X, out,
                                                   (it == N_UNROLL - 1) ? 1 : 0);
    }
}